// Ani_layer_81561428951473
// MI455X (gfx1250) — compile-verified
//
#include <hip/hip_runtime.h>

// Shapes (fixed by the reference)
#define Bn   4
#define IC2  64      // CIN*UM
#define OC2  64      // COUT*UM
#define Hh   128
#define Ww   128
#define KK   576     // IC2 * 9
#define HW   (Hh*Ww)
#define CPAD 68      // padded channel dim (==4 mod 64 -> conflict-free b64 reads)
#define TN   32      // pixels per block (two 16-wide WMMA N-tiles)
#define TCOL 34      // staged columns = TN + 2 halo

typedef __attribute__((ext_vector_type(2))) float v2f;
typedef __attribute__((ext_vector_type(8))) float v8f;

// ---------------------------------------------------------------------------
// Kernel 1: pack weights directly into per-wave WMMA lane order.
// K-order: kp = (dy*3+dx)*64 + c2.
// Wpk[((wv*144 + kt)*32 + lane)*2 + t] = A[o2 = wv*16+(lane&15)]
//                                         [kp = kt*4 + (lane>>4)*2 + t]
// A[o2][kp] = kernel[o2][c2][e=dx][f=dy] = sum_{c,d} params[o,i,c,d] *
//             basis[c,d,dx,dy,g,h2]   (o2=2o+g, c2=2i+h2)
// ---------------------------------------------------------------------------
__global__ __launch_bounds__(256) void build_weights_packed(
    const float* __restrict__ params, const float* __restrict__ basis,
    float* __restrict__ Wpk)
{
    int n = blockIdx.x * blockDim.x + threadIdx.x;
    if (n >= OC2 * KK) return;
    int t    = n & 1;
    int lane = (n >> 1) & 31;
    int rem  = n >> 6;
    int kt   = rem % 144;
    int wv   = rem / 144;

    int o2  = wv * 16 + (lane & 15);
    int kp  = kt * 4 + ((lane >> 4) << 1) + t;
    int tap = kp >> 6;          // 0..8
    int c2  = kp & 63;
    int dy  = tap / 3, dx = tap % 3;
    int o   = o2 >> 1, g  = o2 & 1;
    int i   = c2 >> 1, h2 = c2 & 1;

    float acc = 0.0f;
    const float* pp = params + ((o * 32 + i) * 4) * 4;
    #pragma unroll
    for (int c = 0; c < 4; ++c)
        #pragma unroll
        for (int d = 0; d < 4; ++d)
            acc += pp[c * 4 + d] *
                   basis[((((c * 4 + d) * 3 + dx) * 3 + dy) * 2 + g) * 2 + h2];
    Wpk[n] = acc;
}

// ---------------------------------------------------------------------------
// Kernel 2: S[o2][g'] = sum_{i,e,f} kernel[o2][2i+g'][e][f]
//   = sum_i sum_{c,d} params[o,i,c,d] * (sum_{e,f} basis[c,d,e,f,g,g'])
// 128 threads: tid = o2*2 + g'
// ---------------------------------------------------------------------------
__global__ __launch_bounds__(128) void build_colsums(
    const float* __restrict__ params, const float* __restrict__ basis,
    float* __restrict__ S)
{
    int tid = threadIdx.x;
    int o2 = tid >> 1, gp = tid & 1;
    int o = o2 >> 1, g = o2 & 1;

    float bs[16];
    #pragma unroll
    for (int cd = 0; cd < 16; ++cd) {
        float s = 0.0f;
        #pragma unroll
        for (int ef = 0; ef < 9; ++ef)
            s += basis[((cd * 9 + ef) * 2 + g) * 2 + gp];
        bs[cd] = s;
    }
    float s = 0.0f;
    for (int i = 0; i < 32; ++i) {
        const float* pp = params + ((o * 32 + i) * 4) * 4;
        #pragma unroll
        for (int cd = 0; cd < 16; ++cd)
            s += pp[cd] * bs[cd];
    }
    S[tid] = s;
}

// ---------------------------------------------------------------------------
// Kernel 3: implicit-GEMM conv. Block = 128 threads = 4 waves.
// Block tile = 64 channels x 32 pixels; each wave: 16 ch x 32 px via two
// accumulators, so each A (weight) global_load_b64 feeds TWO
// V_WMMA_F32_16X16X4_F32 (288 WMMAs / 144 A-loads per wave).
// ---------------------------------------------------------------------------
__global__ __launch_bounds__(128) void conv_main(
    const float* __restrict__ x,     // (4, 64, 128, 128)
    const float* __restrict__ Wpk,   // packed weights (64*576)
    const float* __restrict__ S,     // (64, 2)
    const float* __restrict__ bias,  // (64)
    const float* __restrict__ bthr,  // scalar threshold b
    float* __restrict__ out)         // (4, 64, 128, 128)
{
    __shared__ float xt2[3][TCOL][CPAD]; // [row dy][col cc][channel c2], clamped
    __shared__ float avg[2][TN];
    __shared__ float sS[OC2 * 2];
    __shared__ float sbias[OC2];

    const int wt = blockIdx.x;           // w tile (0..3)
    const int h  = blockIdx.y;           // row (0..127)
    const int b  = blockIdx.z;           // batch (0..3)
    const int w0 = wt * TN;
    const int tid = threadIdx.x;

    sS[tid] = S[tid];
    if (tid < OC2) sbias[tid] = bias[tid];

    // Stage clamped 3 x 34 x 64 tile (channel-innermost, padded).
    // Iterated channel-major so global reads are contiguous along ww.
    for (int t = tid; t < IC2 * 3 * TCOL; t += 128) {
        int c2  = t / (3 * TCOL);
        int rem = t % (3 * TCOL);
        int r   = rem / TCOL;
        int cc  = rem % TCOL;
        int hh  = h  - 1 + r;  hh = hh < 0 ? 0 : (hh > Hh - 1 ? Hh - 1 : hh);
        int ww  = w0 - 1 + cc; ww = ww < 0 ? 0 : (ww > Ww - 1 ? Ww - 1 : ww);
        xt2[r][cc][c2] = x[((b * IC2 + c2) * Hh + hh) * Ww + ww];
    }
    __syncthreads();

    // Patch means per um-group g and pixel j (32 channels x 9 taps).
    if (tid < 64) {
        int g = tid & 1, j = tid >> 1;
        float s = 0.0f;
        #pragma unroll
        for (int dy = 0; dy < 3; ++dy)
            #pragma unroll
            for (int dx = 0; dx < 3; ++dx)
                for (int i = 0; i < 32; ++i)
                    s += xt2[dy][j + dx][2 * i + g];
        avg[g][j] = s * (1.0f / 288.0f);
    }
    __syncthreads();

    const int wave  = tid >> 5;
    const int lane  = tid & 31;
    const int base  = wave * 16;
    const int khalf = (lane >> 4) * 2;   // lanes 16-31 hold K+2, K+3
    const int col   = lane & 15;

    // Per-wave packed-weight stream: one coalesced b64 per 2 WMMAs,
    // immediate offsets 0, 256, 512, ... bytes.
    const float* ap = Wpk + ((size_t)(wave * 144) * 32 + lane) * 2;

    v8f acc0 = {};
    v8f acc1 = {};
    #pragma unroll
    for (int tap = 0; tap < 9; ++tap) {
        const int dy = tap / 3, dx = tap % 3;             // compile-time
        const float* bp0 = &xt2[dy][col + dx][khalf];     // pixels 0..15
        const float* bp1 = &xt2[dy][col + 16 + dx][khalf];// pixels 16..31
        #pragma unroll
        for (int t16 = 0; t16 < 16; ++t16) {
            v2f a   = *(const v2f*)(ap  + (tap * 16 + t16) * 64);
            v2f bv0 = *(const v2f*)(bp0 + t16 * 4);
            v2f bv1 = *(const v2f*)(bp1 + t16 * 4);
            acc0 = __builtin_amdgcn_wmma_f32_16x16x4_f32(
                false, a, false, bv0, (short)0, acc0, false, false);
            acc1 = __builtin_amdgcn_wmma_f32_16x16x4_f32(
                false, a, false, bv1, (short)0, acc1, false, false);
        }
    }

    // Epilogue: acc VGPR r = channel (base + r [+8 for lanes 16-31]).
    // (even,odd) VGPR pairs are the (gg=0, gg=1) norm pairs.
    const float thr = bthr[0];
    const int rbase = base + ((lane >> 4) << 3);
    float* outb = out + (size_t)b * OC2 * HW + (size_t)h * Ww + w0;

    #pragma unroll
    for (int px = 0; px < 2; ++px) {
        const int colp = col + 16 * px;
        const float a0 = avg[0][colp];
        const float a1 = avg[1][colp];
        float* outp = outb + colp;
        const v8f& acc = px ? acc1 : acc0;
        #pragma unroll
        for (int pr = 0; pr < 4; ++pr) {
            int r0 = 2 * pr, r1 = 2 * pr + 1;
            int oA = rbase + r0;   // even channel -> gg = 0
            int oB = rbase + r1;   // odd  channel -> gg = 1
            float v0 = acc[r0] - (a0 * sS[oA * 2] + a1 * sS[oA * 2 + 1]) + sbias[oA];
            float v1 = acc[r1] - (a0 * sS[oB * 2] + a1 * sS[oB * 2 + 1]) + sbias[oB];
            float s2  = v0 * v0 + v1 * v1;
            float nrm = __builtin_amdgcn_sqrtf(s2);
            float o0, o1;
            if (nrm - thr <= 0.0f) {
                o0 = 0.0f; o1 = 0.0f;
            } else {
                float inv = __builtin_amdgcn_rsqf(s2);
                o0 = v0 * inv; o1 = v1 * inv;
            }
            o0 += a0;
            o1 += a1;
            outp[(size_t)oA * HW] = o0;
            outp[(size_t)oB * HW] = o1;
        }
    }
}

// ---------------------------------------------------------------------------
extern "C" void kernel_launch(void* const* d_in, const int* in_sizes, int n_in,
                              void* d_out, int out_size, void* d_ws, size_t ws_size,
                              hipStream_t stream)
{
    const float* xx     = (const float*)d_in[0]; // (4, 64, 128, 128)
    const float* params = (const float*)d_in[1]; // (32, 32, 4, 4)
    const float* basis  = (const float*)d_in[2]; // (4, 4, 3, 3, 2, 2)
    const float* bias   = (const float*)d_in[3]; // (1, 32, 2, 1, 1)
    const float* bthr   = (const float*)d_in[4]; // scalar
    float* out = (float*)d_out;

    float* Wpk = (float*)d_ws;           // 64*576 floats, packed
    float* S   = Wpk + OC2 * KK;         // 64*2 floats

    build_weights_packed<<<(OC2 * KK + 255) / 256, 256, 0, stream>>>(params, basis, Wpk);
    build_colsums<<<1, 128, 0, stream>>>(params, basis, S);

    dim3 grid(Ww / TN, Hh, Bn);
    conv_main<<<grid, 128, 0, stream>>>(xx, Wpk, S, bias, bthr, out);
}